// Decoder_GCN_70428873720345
// MI455X (gfx1250) — compile-verified
//
#include <hip/hip_runtime.h>

#define IN_FEAT 256
#define HID 64
#define OUT_F 32

typedef __attribute__((ext_vector_type(2))) float v2f;
typedef __attribute__((ext_vector_type(4))) float v4f;
typedef __attribute__((ext_vector_type(8))) float v8f;

__device__ __forceinline__ void atomic_add_f32(float* p, float v) {
    __hip_atomic_fetch_add(p, v, __ATOMIC_RELAXED, __HIP_MEMORY_SCOPE_AGENT);
}

__device__ __forceinline__ v8f wmma_f32(v2f a, v2f b, v8f c) {
    return __builtin_amdgcn_wmma_f32_16x16x4_f32(false, a, false, b,
                                                 (short)0, c, false, false);
}

// ---------------------------------------------------------------------------
// 1. Fuse fc1 and gcn linear:  Wc[k][o] = sum_h fc1_w[h][k] * gcn_w[h][o]
//    bc[o] = sum_h fc1_b[h] * gcn_w[h][o]
// ---------------------------------------------------------------------------
__global__ void wc_kernel(const float* __restrict__ fc1_w,
                          const float* __restrict__ fc1_b,
                          const float* __restrict__ gcn_w,
                          float* __restrict__ wc, float* __restrict__ bc) {
    int id = blockIdx.x * blockDim.x + threadIdx.x;
    if (id < IN_FEAT * OUT_F) {
        int k = id / OUT_F, o = id % OUT_F;
        float s = 0.f;
        #pragma unroll 8
        for (int h = 0; h < HID; ++h)
            s += fc1_w[h * IN_FEAT + k] * gcn_w[h * OUT_F + o];
        wc[id] = s;
    }
    if (id < OUT_F) {
        float s = 0.f;
        for (int h = 0; h < HID; ++h)
            s += fc1_b[h] * gcn_w[h * OUT_F + id];
        bc[id] = s;
    }
}

// ---------------------------------------------------------------------------
// 2-4. degree / rsqrt pipeline (deg includes the self-loop: init 1.0)
// ---------------------------------------------------------------------------
__global__ void deg_init_kernel(float* __restrict__ deg, int n) {
    int i = blockIdx.x * blockDim.x + threadIdx.x;
    if (i < n) deg[i] = 1.0f;
}

__global__ void deg_scatter_kernel(const long long* __restrict__ dst,
                                   float* __restrict__ deg, long long e) {
    long long i = (long long)blockIdx.x * blockDim.x + threadIdx.x;
    if (i < e) atomic_add_f32(&deg[dst[i]], 1.0f);
}

__global__ void dinv_kernel(float* __restrict__ deg, int n) {
    int i = blockIdx.x * blockDim.x + threadIdx.x;
    if (i < n) deg[i] = __frsqrt_rn(deg[i]);   // deg >= 1 always
}

// ---------------------------------------------------------------------------
// 5. hw = X @ Wc + bc  via V_WMMA_F32_16X16X4_F32.
//
//    Wc is pre-swizzled into LDS in *fragment order*: for K-block j (8 K
//    values) the four WMMA B operands (2 K-quads x 2 N-tiles) are stored as
//    one contiguous v2f per lane, so the inner loop is a straight
//    ds_load_b64 into an even VGPR pair (no repacking).
//
//    Effective K grouping per 8-block (from the single b128 A load with
//    kb = 8j + 4*(lane>=16)):
//      frag0 = K {8j,8j+1 | 8j+4,8j+5},  frag1 = K {8j+2,8j+3 | 8j+6,8j+7}
//    The same permutation is baked into the B swizzle, so sums are exact.
//
//    Each wave owns M=32 rows (two 16-row tiles) sharing the B fragments:
//    per iteration 2x global b128 + 4x ds b64 + 8x WMMA.
// ---------------------------------------------------------------------------
__global__ __launch_bounds__(256) void gemm_hw_kernel(
        const float* __restrict__ x, const float* __restrict__ wc,
        const float* __restrict__ bc, float* __restrict__ hw, int n_rows) {
    __shared__ v2f lds_b[(IN_FEAT / 8) * 4 * 32];   // 32 KB
    float* lds_bf = (float*)lds_b;

    // swizzle Wc[k][o] -> lds_b[j][frag][t][lane][pos]
    for (int i = threadIdx.x; i < IN_FEAT * OUT_F; i += blockDim.x) {
        int k = i >> 5;            // k index (o stride 1)
        int o = i & 31;
        int j = k >> 3, r = k & 7;
        int frag = (r >> 1) & 1;   // 0: K r=0,1|4,5   1: K r=2,3|6,7
        int pos  = r & 1;
        int lane = (o & 15) + ((r >> 2) << 4);
        int t    = o >> 4;
        lds_bf[((((j * 2 + frag) * 2 + t) * 32) + lane) * 2 + pos] = wc[i];
    }
    __syncthreads();

    const int lane = threadIdx.x & 31;
    const int wave = threadIdx.x >> 5;
    const int wavesPerBlock = blockDim.x >> 5;
    const int half = lane >> 4;
    const int lm   = lane & 15;

    const float bias0 = bc[lm];
    const float bias1 = bc[lm + 16];

    const int numTiles = (n_rows + 31) >> 5;   // 32-row tiles
    for (int tile = blockIdx.x * wavesPerBlock + wave; tile < numTiles;
         tile += gridDim.x * wavesPerBlock) {
        const int row0 = tile * 32;
        const int arowA = row0 + lm;
        const int arowB = row0 + 16 + lm;
        const float* xrowA =
            x + (size_t)(arowA < n_rows ? arowA : 0) * IN_FEAT;
        const float* xrowB =
            x + (size_t)(arowB < n_rows ? arowB : 0) * IN_FEAT;

        v8f cA0 = {0.f, 0.f, 0.f, 0.f, 0.f, 0.f, 0.f, 0.f};
        v8f cA1 = cA0, cB0 = cA0, cB1 = cA0;

        #pragma unroll 2
        for (int j = 0; j < IN_FEAT / 8; ++j) {
            const int kb = 8 * j + 4 * half;
            const v4f aA = __builtin_nontemporal_load((const v4f*)(xrowA + kb));
            const v4f aB = __builtin_nontemporal_load((const v4f*)(xrowB + kb));

            const v2f* bj = lds_b + j * 4 * 32 + lane;
            v2f b00 = bj[0];    // frag0, N-tile 0
            v2f b01 = bj[32];   // frag0, N-tile 1
            v2f b10 = bj[64];   // frag1, N-tile 0
            v2f b11 = bj[96];   // frag1, N-tile 1

            v2f a0A = {aA.x, aA.y}, a1A = {aA.z, aA.w};
            v2f a0B = {aB.x, aB.y}, a1B = {aB.z, aB.w};

            cA0 = wmma_f32(a0A, b00, cA0);
            cA1 = wmma_f32(a0A, b01, cA1);
            cA0 = wmma_f32(a1A, b10, cA0);
            cA1 = wmma_f32(a1A, b11, cA1);
            cB0 = wmma_f32(a0B, b00, cB0);
            cB1 = wmma_f32(a0B, b01, cB1);
            cB0 = wmma_f32(a1B, b10, cB0);
            cB1 = wmma_f32(a1B, b11, cB1);
        }

        // C/D layout: VGPR r -> M = r + 8*half, N = lm (t0) / lm+16 (t1)
        #pragma unroll
        for (int r = 0; r < 8; ++r) {
            const int orowA = row0 + r + 8 * half;
            const int orowB = orowA + 16;
            if (orowA < n_rows) {
                hw[(size_t)orowA * OUT_F + lm]      = cA0[r] + bias0;
                hw[(size_t)orowA * OUT_F + lm + 16] = cA1[r] + bias1;
            }
            if (orowB < n_rows) {
                hw[(size_t)orowB * OUT_F + lm]      = cB0[r] + bias0;
                hw[(size_t)orowB * OUT_F + lm + 16] = cB1[r] + bias1;
            }
        }
    }
}

// ---------------------------------------------------------------------------
// 6. out[i,f] = gcn_b[f] + hw[i,f] * dinv[i]^2   (self-loop contribution)
// ---------------------------------------------------------------------------
__global__ void out_init_kernel(const float* __restrict__ hw,
                                const float* __restrict__ dinv,
                                const float* __restrict__ gcn_b,
                                float* __restrict__ out, int n) {
    long long id = (long long)blockIdx.x * blockDim.x + threadIdx.x;
    long long i = id >> 5;
    int f = (int)(id & 31);
    if (i < n) {
        float di = dinv[i];
        out[id] = gcn_b[f] + hw[id] * di * di;
    }
}

// ---------------------------------------------------------------------------
// 7. Edge scatter: one wave per edge, lane = feature.
//    out[dst,f] += hw[src,f] * dinv[src]*dinv[dst]
// ---------------------------------------------------------------------------
__global__ void scatter_kernel(const long long* __restrict__ src,
                               const long long* __restrict__ dst,
                               const float* __restrict__ dinv,
                               const float* __restrict__ hw,
                               float* __restrict__ out, long long e) {
    long long id = (long long)blockIdx.x * blockDim.x + threadIdx.x;
    long long edge = id >> 5;
    int f = (int)(id & 31);
    if (edge < e) {
        long long s = src[edge];
        long long d = dst[edge];
        float norm = dinv[s] * dinv[d];
        atomic_add_f32(&out[d * OUT_F + f], hw[s * OUT_F + f] * norm);
    }
}

// ---------------------------------------------------------------------------
extern "C" void kernel_launch(void* const* d_in, const int* in_sizes, int n_in,
                              void* d_out, int out_size, void* d_ws, size_t ws_size,
                              hipStream_t stream) {
    const float*     x      = (const float*)d_in[0];
    const long long* ei     = (const long long*)d_in[1];   // int64 [2, E]
    const float*     fc1_w  = (const float*)d_in[2];
    const float*     fc1_b  = (const float*)d_in[3];
    const float*     gcn_w  = (const float*)d_in[4];
    const float*     gcn_b  = (const float*)d_in[5];
    float*           out    = (float*)d_out;

    const int       n = in_sizes[0] / IN_FEAT;
    const long long e = in_sizes[1] / 2;
    const long long* srcIdx = ei;
    const long long* dstIdx = ei + e;

    // workspace layout (floats)
    float* ws  = (float*)d_ws;
    float* wc  = ws;                       // 256*32
    float* bc  = wc + IN_FEAT * OUT_F;     // 32
    float* deg = bc + OUT_F;               // n (reused as dinv)
    float* hw  = deg + n;                  // n*32

    // 1. fused weight
    wc_kernel<<<(IN_FEAT * OUT_F + 255) / 256, 256, 0, stream>>>(fc1_w, fc1_b,
                                                                 gcn_w, wc, bc);
    // 2-4. degree -> dinv
    deg_init_kernel<<<(n + 255) / 256, 256, 0, stream>>>(deg, n);
    deg_scatter_kernel<<<(unsigned)((e + 255) / 256), 256, 0, stream>>>(dstIdx, deg, e);
    dinv_kernel<<<(n + 255) / 256, 256, 0, stream>>>(deg, n);

    // 5. WMMA GEMM: hw = x @ wc + bc   (32-row tiles, 8 waves per block)
    const int tiles  = (n + 31) / 32;
    const int blocks = (tiles + 7) / 8;
    gemm_hw_kernel<<<blocks, 256, 0, stream>>>(x, wc, bc, hw, n);

    // 6. self-loop + bias init
    out_init_kernel<<<(unsigned)(((long long)n * OUT_F + 255) / 256), 256, 0,
                      stream>>>(hw, deg, gcn_b, out, n);

    // 7. edge scatter (one wave per edge)
    const long long sthreads = e * 32;
    scatter_kernel<<<(unsigned)((sthreads + 255) / 256), 256, 0, stream>>>(
        srcIdx, dstIdx, deg, hw, out, e);
}